// Net_28544352649361
// MI455X (gfx1250) — compile-verified
//
#include <hip/hip_runtime.h>
#include <hip/hip_bf16.h>

typedef __attribute__((ext_vector_type(2))) float v2f;
typedef __attribute__((ext_vector_type(8))) float v8f;

// ---------------------------------------------------------------------------
// Kernel A: binary-search segment boundaries in the sorted segment_ids array.
// bounds[s] = first index i with segment_ids[i] >= s, for s in [0, n_seg].
// ---------------------------------------------------------------------------
__global__ void seg_bounds_kernel(const int* __restrict__ seg_ids, int T,
                                  int n_seg, int* __restrict__ bounds) {
    int s = blockIdx.x * blockDim.x + threadIdx.x;
    if (s > n_seg) return;
    int lo = 0, hi = T;
    while (lo < hi) {
        int mid = (lo + hi) >> 1;
        if (seg_ids[mid] < s) lo = mid + 1; else hi = mid;
    }
    bounds[s] = lo;
}

// ---------------------------------------------------------------------------
// Kernel B: one wave (32 lanes) per segment. For each token in the segment,
// the wave loads the 128-float embedding row (lane l -> float4 at l*4),
// accumulates a per-lane partial, then cross-lane reduces to the segment's
// scalar mean. Fully coalesced (4 whole cachelines per token), no atomics.
// ---------------------------------------------------------------------------
__global__ void seg_mean_kernel(const int* __restrict__ tokens,
                                const float* __restrict__ wv,   // [V,128]
                                const int* __restrict__ bounds,
                                int n_seg,
                                float* __restrict__ sv)         // [n_seg]
{
    int wave = threadIdx.x >> 5;
    int lane = threadIdx.x & 31;
    int s = blockIdx.x * (blockDim.x >> 5) + wave;
    if (s >= n_seg) return;

    int start = bounds[s];
    int end   = bounds[s + 1];

    float acc = 0.0f;
    for (int t = start; t < end; ++t) {
        int tok = tokens[t];                          // uniform across wave
        const float4* row = (const float4*)(wv + (size_t)tok * 128);
        float4 v = row[lane];                         // global_load_b128
        acc += (v.x + v.y) + (v.z + v.w);
    }
    // wave32 tree reduction
    #pragma unroll
    for (int off = 16; off > 0; off >>= 1)
        acc += __shfl_down(acc, off, 32);

    if (lane == 0) {
        float len = (float)(end - start);
        sv[s] = acc / fmaxf(len, 1.0f);
    }
}

// ---------------------------------------------------------------------------
// Kernel C: out = sv(broadcast) @ W^T + b via V_WMMA_F32_16X16X4_F32.
// One wave per 16x16 output tile; 8 waves/block cover all 128 labels for a
// 16-sentence row block. K-loop: 32 steps of K=4, f32 accumulate (exact).
//
// Layouts (CDNA5 ISA 7.12.2, 32-bit):
//   A 16x4 : lanes 0-15 M=lane  (VGPR0=K0, VGPR1=K1); lanes 16-31 M=lane-16
//            (VGPR0=K2, VGPR1=K3). sv rows are constant -> both regs = s.
//   B 4x16 : N on lanes (mirror of A): lane half selects K pair {0,1}/{2,3}.
//   C/D    : VGPR r, lane L -> M = r + 8*(L>=16), N = L&15.
// ---------------------------------------------------------------------------
__global__ void __launch_bounds__(256)
wmma_out_kernel(const float* __restrict__ sv,     // [n_sent]
                const float* __restrict__ W,      // [n_labels, 128]
                const float* __restrict__ bias,   // [n_labels]
                float* __restrict__ out,          // [n_sent, n_labels]
                int n_labels)
{
    const int wave = threadIdx.x >> 5;           // 0..7 -> column tile
    const int lane = threadIdx.x & 31;
    const int l15  = lane & 15;
    const int half = lane >> 4;                  // 0 or 1

    const int row_base = blockIdx.x * 16;
    const int col      = (wave << 4) + l15;      // this lane's label column

    // A operand: constant per sentence row, same for every K step.
    float s = sv[row_base + l15];
    v2f a; a.x = s; a.y = s;

    // C operand seeded with bias: C[m, n] = b[n] for all m.
    float bv = bias[col];
    v8f c;
    #pragma unroll
    for (int r = 0; r < 8; ++r) c[r] = bv;

    // B operand: B[k, n] = W[n, k]; this lane supplies K = kbase+2*half (+1).
    const float* wrow = W + (size_t)col * 128 + 2 * half;

    #pragma unroll
    for (int k = 0; k < 128; k += 4) {
        float2 wp = *(const float2*)(wrow + k);  // global_load_b64
        v2f b; b.x = wp.x; b.y = wp.y;
        c = __builtin_amdgcn_wmma_f32_16x16x4_f32(
                /*neg_a=*/false, a, /*neg_b=*/false, b,
                /*c_mod=*/(short)0, c, /*reuse_a=*/false, /*reuse_b=*/false);
    }

    // Store D: lane (r, L) -> out[row_base + r + 8*half, col]
    #pragma unroll
    for (int r = 0; r < 8; ++r) {
        int m = r + 8 * half;
        out[(size_t)(row_base + m) * n_labels + col] = c[r];
    }
}

// ---------------------------------------------------------------------------
extern "C" void kernel_launch(void* const* d_in, const int* in_sizes, int n_in,
                              void* d_out, int out_size, void* d_ws, size_t ws_size,
                              hipStream_t stream) {
    const int*   tokens   = (const int*)  d_in[0];
    const int*   seg_ids  = (const int*)  d_in[1];
    const float* wv       = (const float*)d_in[2];
    const float* hidden_w = (const float*)d_in[3];
    const float* hidden_b = (const float*)d_in[4];
    float*       out      = (float*)d_out;

    const int T        = in_sizes[0];              // 1048576 tokens
    const int n_labels = in_sizes[4];              // 128
    const int n_sent   = out_size / n_labels;      // 16384

    // Workspace: bounds[n_sent+1] ints, then sv[n_sent] floats (256B aligned).
    int*   bounds = (int*)d_ws;
    size_t off    = (((size_t)(n_sent + 1) * sizeof(int)) + 255) & ~(size_t)255;
    float* sv     = (float*)((char*)d_ws + off);

    // A: segment boundaries via binary search (16385 threads).
    {
        int threads = 256;
        int blocks  = (n_sent + 1 + threads - 1) / threads;
        seg_bounds_kernel<<<blocks, threads, 0, stream>>>(seg_ids, T, n_sent, bounds);
    }

    // B: per-segment scalar mean, one wave per segment, 8 waves per block.
    {
        int threads = 256;
        int wavesPerBlock = threads / 32;
        int blocks = (n_sent + wavesPerBlock - 1) / wavesPerBlock;
        seg_mean_kernel<<<blocks, threads, 0, stream>>>(tokens, wv, bounds, n_sent, sv);
    }

    // C: WMMA f32 GEMM  out = sv_broadcast @ W^T + b.
    {
        int blocks = n_sent / 16;                  // 1024 row tiles
        wmma_out_kernel<<<blocks, 256, 0, stream>>>(sv, hidden_w, hidden_b, out, n_labels);
    }
}